// KKTLoss_60748017435017
// MI455X (gfx1250) — compile-verified
//
#include <hip/hip_runtime.h>

// ---------------------------------------------------------------------------
// KKT residual loss for batched sparse LPs (B=64, M=4096, N=8192, ~10.24M nnz)
//
// Pass 1: zero Ax[B*M], AtLam[B*N] in workspace, zero out[0]
// Pass 2: fused COO scatter:  Ax[r] += v*x[c];  AtLam[c] += v*lam[r]
//         (reads each COO triple ONCE for both products; float4/int4 loads)
// Pass 3: elementwise residuals + global weighted sum.
//         Wave-level reduction uses V_WMMA_F32_16X16X4_F32 (A = partials,
//         B = ones -> D rows hold exact f32 wave sums), then LDS ds_add_f32
//         across waves, then one global atomic per block.
// ---------------------------------------------------------------------------

typedef __attribute__((ext_vector_type(2))) float v2f;
typedef __attribute__((ext_vector_type(8))) float v8f;

__global__ void kkt_zero_kernel(float* __restrict__ acc, float* __restrict__ out, int n) {
    int stride = gridDim.x * blockDim.x;
    for (int i = blockIdx.x * blockDim.x + threadIdx.x; i < n; i += stride)
        acc[i] = 0.0f;
    if (blockIdx.x == 0 && threadIdx.x == 0)
        out[0] = 0.0f;
}

__global__ void kkt_coo_kernel(const float* __restrict__ vals,
                               const int*   __restrict__ rows,
                               const int*   __restrict__ cols,
                               const float* __restrict__ x,
                               const float* __restrict__ lam,
                               float* __restrict__ Ax,
                               float* __restrict__ AtL,
                               int nnz4, int nnz) {
    int stride = gridDim.x * blockDim.x;
    int tid = blockIdx.x * blockDim.x + threadIdx.x;

    // vectorized main body: 4 nonzeros per iteration via B128 loads
    const float4* v4 = (const float4*)vals;
    const int4*   r4 = (const int4*)rows;
    const int4*   c4 = (const int4*)cols;
    for (int i = tid; i < nnz4; i += stride) {
        float4 v = v4[i];
        int4   r = r4[i];
        int4   c = c4[i];
        atomicAdd(&Ax[r.x],  v.x * x[c.x]);
        atomicAdd(&AtL[c.x], v.x * lam[r.x]);
        atomicAdd(&Ax[r.y],  v.y * x[c.y]);
        atomicAdd(&AtL[c.y], v.y * lam[r.y]);
        atomicAdd(&Ax[r.z],  v.z * x[c.z]);
        atomicAdd(&AtL[c.z], v.z * lam[r.z]);
        atomicAdd(&Ax[r.w],  v.w * x[c.w]);
        atomicAdd(&AtL[c.w], v.w * lam[r.w]);
    }
    // scalar tail (nnz not divisible by 4)
    for (int i = nnz4 * 4 + tid; i < nnz; i += stride) {
        float v = vals[i];
        int   r = rows[i];
        int   c = cols[i];
        atomicAdd(&Ax[r],  v * x[c]);
        atomicAdd(&AtL[c], v * lam[r]);
    }
}

// Exact-f32 wave32 sum via V_WMMA_F32_16X16X4_F32, then LDS + global atomics.
// A matrix (16x4 f32, 2 VGPRs): lanes 0-15 supply K=0,1; lanes 16-31 K=2,3.
// With a = {p, 0}: A[m][0] = p_m, A[m][2] = p_{m+16}. B = ones(4x16) gives
// D[m][n] = p_m + p_{m+16}. Lane 0 holds rows 0-7 in d[0..7], lane 16 rows
// 8-15, so (lane0 sum) + (lane16 sum) = full wave sum, all in f32 FMA.
__device__ __forceinline__ void kkt_block_reduce_atomic(float p, float* __restrict__ out) {
    __shared__ float bacc;
    if (threadIdx.x == 0) bacc = 0.0f;
    __syncthreads();
    unsigned lane = threadIdx.x & 31u;
#if __has_builtin(__builtin_amdgcn_wmma_f32_16x16x4_f32)
    v2f a; a[0] = p;    a[1] = 0.0f;
    v2f b; b[0] = 1.0f; b[1] = 1.0f;
    v8f c = {};
    v8f d = __builtin_amdgcn_wmma_f32_16x16x4_f32(
        /*neg_a=*/false, a, /*neg_b=*/false, b,
        /*c_mod=*/(short)0, c, /*reuse_a=*/false, /*reuse_b=*/false);
    float w = 0.0f;
#pragma unroll
    for (int i = 0; i < 8; ++i) w += d[i];
    if (lane == 0u || lane == 16u) atomicAdd(&bacc, w);
#else
    float w = p;
#pragma unroll
    for (int off = 16; off > 0; off >>= 1) w += __shfl_xor(w, off, 32);
    if (lane == 0u) atomicAdd(&bacc, w);
#endif
    __syncthreads();
    if (threadIdx.x == 0) atomicAdd(out, bacc);
}

__global__ void kkt_loss_kernel(const float* __restrict__ Ax,
                                const float* __restrict__ AtL,
                                const float* __restrict__ bvec,
                                const float* __restrict__ cvec,
                                const float* __restrict__ x,
                                const float* __restrict__ lam,
                                float* __restrict__ out,
                                int bm, int bn) {
    // weights: per-instance /den then mean over B folds into flat constants:
    //   row & col primal/dual/comp terms: W/( (M+N)*B ) = W/(bm+bn)
    //   stationarity (cols only):         W_STAT/( N*B ) = W_STAT/bn
    const float wPD = 0.1f / (float)(bm + bn);   // W_PRIMAL == W_DUAL == 0.1
    const float wS  = 0.6f / (float)bn;
    const float wC  = 0.2f / (float)(bm + bn);

    int stride = gridDim.x * blockDim.x;
    int total = bm + bn;
    float p = 0.0f;
    for (int i = blockIdx.x * blockDim.x + threadIdx.x; i < total; i += stride) {
        if (i < bm) {
            // row space: primal residual, dual feas on lambda, comp (lam*axmb)
            float axmb = Ax[i] - bvec[i];
            float l    = lam[i];
            float rp = axmb > 0.0f ? axmb : 0.0f;    // relu(Ax-b)
            float rl = l < 0.0f ? -l : 0.0f;         // relu(-lam)
            float cm = l * axmb;
            p += wPD * (rp * rp + rl * rl) + wC * (cm * cm);
        } else {
            int j = i - bm;
            // col space: mu = relu(c + AtLam); stat = (c+AtLam) - mu = min(.,0)
            float t  = cvec[j] + AtL[j];
            float mu = t > 0.0f ? t : 0.0f;
            float st = t - mu;                        // stationarity residual
            float xv = x[j];
            float rx = xv < 0.0f ? -xv : 0.0f;        // relu(-x)
            float cm = mu * xv;
            // relu(-mu) == 0 identically since mu >= 0
            p += wPD * (rx * rx) + wS * (st * st) + wC * (cm * cm);
        }
    }
    kkt_block_reduce_atomic(p, out);
}

extern "C" void kernel_launch(void* const* d_in, const int* in_sizes, int n_in,
                              void* d_out, int out_size, void* d_ws, size_t ws_size,
                              hipStream_t stream) {
    const float* x_hat   = (const float*)d_in[0];
    const float* lam_hat = (const float*)d_in[1];
    const float* a_vals  = (const float*)d_in[2];
    const int*   a_rows  = (const int*)d_in[3];
    const int*   a_cols  = (const int*)d_in[4];
    const float* b_pad   = (const float*)d_in[5];
    const float* c_pad   = (const float*)d_in[6];
    float* out = (float*)d_out;

    const int bn  = in_sizes[0];   // B*N
    const int bm  = in_sizes[1];   // B*M
    const int nnz = in_sizes[2];

    float* Ax  = (float*)d_ws;     // bm floats
    float* AtL = Ax + bm;          // bn floats (Ax/AtL contiguous)

    // Pass 1: zero accumulators (contiguous bm+bn floats) and output scalar
    kkt_zero_kernel<<<768, 256, 0, stream>>>(Ax, out, bm + bn);

    // Pass 2: fused COO double scatter
    int nnz4 = nnz >> 2;
    int cb = (nnz4 + 255) / 256;
    if (cb > 4096) cb = 4096;
    if (cb < 1)    cb = 1;
    kkt_coo_kernel<<<cb, 256, 0, stream>>>(a_vals, a_rows, a_cols,
                                           x_hat, lam_hat, Ax, AtL, nnz4, nnz);

    // Pass 3: residuals + global weighted reduction
    kkt_loss_kernel<<<768, 256, 0, stream>>>(Ax, AtL, b_pad, c_pad,
                                             x_hat, lam_hat, out, bm, bn);
}